// GATHead_76982993814214
// MI455X (gfx1250) — compile-verified
//
#include <hip/hip_runtime.h>
#include <hip/hip_bf16.h>
#include <stdint.h>

// ---------------- problem constants (match reference) ----------------
#define NN       50000
#define EE       800000
#define IN_DIM   1024
#define HIDC     64
#define HEADS1   4
#define NCLASS   2
#define NGRAPHS  8
#define NEGSLOPE 0.2f
#define EPSV     1e-16f

typedef float v2f __attribute__((ext_vector_type(2)));
typedef float v8f __attribute__((ext_vector_type(8)));

// ---------------- helpers ----------------
__device__ __forceinline__ void edge_sd(const long long* __restrict__ ei, int e,
                                        int nE, int& s, int& d) {
  if (e < nE) { s = (int)ei[e]; d = (int)ei[(size_t)nE + e]; }
  else        { s = e - nE; d = s; }  // appended self-loops
}

// monotone float -> uint key so unsigned atomicMax == float max
__device__ __forceinline__ unsigned fkey(float f) {
  unsigned u = __float_as_uint(f);
  return (u & 0x80000000u) ? ~u : (u | 0x80000000u);
}
__device__ __forceinline__ float fdecode(unsigned u) {
  unsigned b = (u & 0x80000000u) ? (u ^ 0x80000000u) : ~u;
  return __uint_as_float(b);
}

// ---------------- WMMA f32 GEMM: C[M,TN] = A[M,TK] @ B[TK,TN] --------------
// Compile-time TN/TK so every B access is base + immediate offset (no 64-bit
// per-iteration address math) and the k-loop strength-reduces to pointer adds.
// blockDim.x = 128 (4 waves). Each wave computes TWO adjacent 16x16 tiles,
// reusing one A fragment across two V_WMMA_F32_16X16X4_F32 per k-step.
// grid.x = M/16 (M multiple of 16); grid.y = ceil(TN/128). Waves whose column
// range falls outside TN exit wave-uniformly, so EXEC is all-ones at every
// WMMA as the ISA requires.
template <int TN, int TK>
__global__ void wmma_gemm_f32_kernel(const float* __restrict__ A,
                                     const float* __restrict__ B,
                                     float* __restrict__ C) {
  const int lane = threadIdx.x & 31;
  const int wave = threadIdx.x >> 5;
  const int half = lane >> 4;   // K-pair select (A: K+2*half, B: row K+2*half)
  const int lm   = lane & 15;   // M index for A frag, N index for B frag
  const int mbase = blockIdx.x * 16;
  const int nbase = blockIdx.y * 128 + wave * 32;   // two 16-wide tiles
  if (nbase >= TN) return;      // wave-uniform exit (e.g. TN=64 -> waves 2,3)

  v8f acc0 = {};
  v8f acc1 = {};
  const float* __restrict__ arow = A + (size_t)(mbase + lm) * TK;
  const float* __restrict__ bcol = B + nbase + lm;  // + k*TN folds to imm offs
#pragma unroll 4
  for (int k = 0; k < TK; k += 4) {
    const int ka = k + 2 * half;
    v2f a, b0, b1;
    a = *(const v2f*)(arow + ka);                   // one 8-byte load
    b0.x = bcol[(size_t)ka * TN];
    b0.y = bcol[(size_t)(ka + 1) * TN];
    b1.x = bcol[(size_t)ka * TN + 16];
    b1.y = bcol[(size_t)(ka + 1) * TN + 16];
    acc0 = __builtin_amdgcn_wmma_f32_16x16x4_f32(
        false, a, false, b0, (short)0, acc0, false, false);
    acc1 = __builtin_amdgcn_wmma_f32_16x16x4_f32(
        false, a, false, b1, (short)0, acc1, false, false);
  }
  // C/D layout: VGPR i, lanes 0-15 -> M=i, lanes 16-31 -> M=i+8; N = lane%16
  float* __restrict__ crow = C + (size_t)(mbase + 8 * half) * TN + nbase + lm;
#pragma unroll
  for (int i = 0; i < 8; ++i) {
    crow[(size_t)i * TN]      = acc0[i];
    crow[(size_t)i * TN + 16] = acc1[i];
  }
}

// ---------------- zero fill ----------------
__global__ void zero_kernel(float* __restrict__ p, size_t n) {
  size_t i = (size_t)blockIdx.x * blockDim.x + threadIdx.x;
  if (i < n) p[i] = 0.0f;
}

// ---------------- per-node attention logits ----------------
// asrc[n,h] = dot(h[n,h,:], att_src[h,:]); adst likewise. 64 ch per head.
__global__ void attn_node_kernel(const float* __restrict__ h,
                                 const float* __restrict__ att_src,
                                 const float* __restrict__ att_dst,
                                 float* __restrict__ asrc,
                                 float* __restrict__ adst,
                                 int n_nodes, int H) {
  int idx = blockIdx.x * blockDim.x + threadIdx.x;
  if (idx >= n_nodes * H) return;
  int n = idx / H, hh = idx - n * H;
  const float* hv = h + ((size_t)n * H + hh) * HIDC;
  const float* as = att_src + hh * HIDC;
  const float* ad = att_dst + hh * HIDC;
  float s = 0.f, d = 0.f;
#pragma unroll 8
  for (int c = 0; c < HIDC; ++c) { float v = hv[c]; s += v * as[c]; d += v * ad[c]; }
  asrc[idx] = s;
  adst[idx] = d;
}

// ---------------- segment max over edges (keyed unsigned atomicMax) --------
__global__ void edge_max_kernel(const float* __restrict__ asrc,
                                const float* __restrict__ adst,
                                const long long* __restrict__ ei,
                                unsigned* __restrict__ emax,
                                int nE, int n_nodes, int H) {
  int idx = blockIdx.x * blockDim.x + threadIdx.x;
  int tot = (nE + n_nodes) * H;
  if (idx >= tot) return;
  int e = idx / H, hh = idx - e * H;
  int s, d; edge_sd(ei, e, nE, s, d);
  float v = asrc[(size_t)s * H + hh] + adst[(size_t)d * H + hh];
  v = (v > 0.f) ? v : NEGSLOPE * v;                 // LeakyReLU
  atomicMax(&emax[(size_t)d * H + hh], fkey(v));
}

// ---------------- exp(e - max) + segment sum; stash per-edge weight -------
__global__ void edge_exp_kernel(const float* __restrict__ asrc,
                                const float* __restrict__ adst,
                                const long long* __restrict__ ei,
                                const unsigned* __restrict__ emax,
                                float* __restrict__ ealpha,
                                float* __restrict__ denom,
                                int nE, int n_nodes, int H) {
  int idx = blockIdx.x * blockDim.x + threadIdx.x;
  int tot = (nE + n_nodes) * H;
  if (idx >= tot) return;
  int e = idx / H, hh = idx - e * H;
  int s, d; edge_sd(ei, e, nE, s, d);
  float v = asrc[(size_t)s * H + hh] + adst[(size_t)d * H + hh];
  v = (v > 0.f) ? v : NEGSLOPE * v;
  float m = fdecode(emax[(size_t)d * H + hh]);      // every dst has a self-loop
  float w = expf(v - m);
  ealpha[idx] = w;
  atomicAdd(&denom[(size_t)d * H + hh], w);
}

// ---------------- alpha-weighted scatter-add of messages ------------------
// one block per edge; blockDim = H*64 threads; thread t -> head t/64, ch t%64
__global__ void scatter_kernel(const float* __restrict__ hsrc,
                               const float* __restrict__ ealpha,
                               const float* __restrict__ denom,
                               const long long* __restrict__ ei,
                               float* __restrict__ out,
                               int nE, int n_nodes, int H) {
  int e = blockIdx.x;
  int t = threadIdx.x;
  int s, d; edge_sd(ei, e, nE, s, d);
  int hh = t >> 6;
  float alpha = ealpha[(size_t)e * H + hh] /
                (denom[(size_t)d * H + hh] + EPSV);
  int F = H * HIDC;
  float msg = hsrc[(size_t)s * F + t] * alpha;
  atomicAdd(&out[(size_t)d * F + t], msg);
}

// ---------------- bias + ReLU (in place) ----------------
__global__ void bias_relu_kernel(float* __restrict__ p,
                                 const float* __restrict__ bias,
                                 int ld, size_t n) {
  size_t i = (size_t)blockIdx.x * blockDim.x + threadIdx.x;
  if (i >= n) return;
  float v = p[i] + bias[i % ld];
  p[i] = v > 0.f ? v : 0.f;
}

// ---------------- global mean pool ----------------
__global__ void pool_accum_kernel(const float* __restrict__ h,
                                  const long long* __restrict__ batch,
                                  float* __restrict__ sums, int n_nodes) {
  int idx = blockIdx.x * blockDim.x + threadIdx.x;
  if (idx >= n_nodes * HIDC) return;
  int n = idx / HIDC, c = idx - n * HIDC;
  int b = (int)batch[n];
  atomicAdd(&sums[b * HIDC + c], h[idx]);
}
__global__ void pool_count_kernel(const long long* __restrict__ batch,
                                  float* __restrict__ cnts, int n_nodes) {
  int n = blockIdx.x * blockDim.x + threadIdx.x;
  if (n >= n_nodes) return;
  atomicAdd(&cnts[(int)batch[n]], 1.0f);
}
__global__ void final_lin_kernel(const float* __restrict__ sums,
                                 const float* __restrict__ cnts,
                                 const float* __restrict__ lin_w,
                                 const float* __restrict__ lin_b,
                                 float* __restrict__ out) {
  int t = threadIdx.x;
  if (t >= NGRAPHS * NCLASS) return;
  int g = t / NCLASS, cls = t - g * NCLASS;
  float cnt = cnts[g]; cnt = cnt > 1.0f ? cnt : 1.0f;
  float acc = lin_b[cls];
  for (int c = 0; c < HIDC; ++c)
    acc += (sums[g * HIDC + c] / cnt) * lin_w[c * NCLASS + cls];
  out[g * NCLASS + cls] = acc;
}

// ---------------- host orchestration ----------------
extern "C" void kernel_launch(void* const* d_in, const int* in_sizes, int n_in,
                              void* d_out, int out_size, void* d_ws, size_t ws_size,
                              hipStream_t stream) {
  const float*      x        = (const float*)d_in[0];
  const long long*  ei       = (const long long*)d_in[1];   // int64 [2,E]
  const long long*  batch    = (const long long*)d_in[2];   // int64 [N]
  const float*      W1       = (const float*)d_in[3];
  const float*      att_src1 = (const float*)d_in[4];
  const float*      att_dst1 = (const float*)d_in[5];
  const float*      b1       = (const float*)d_in[6];
  const float*      W2       = (const float*)d_in[7];
  const float*      att_src2 = (const float*)d_in[8];
  const float*      att_dst2 = (const float*)d_in[9];
  const float*      b2       = (const float*)d_in[10];
  const float*      lin_w    = (const float*)d_in[11];
  const float*      lin_b    = (const float*)d_in[12];
  float*            out      = (float*)d_out;

  // -------- workspace layout (aliased across layers), ~120 MB --------
  char* ws = (char*)d_ws;
  size_t o = 0;
  float* h1     = (float*)(ws + o); o += (size_t)NN * 256 * 4;      // h2 aliases
  float* acc1   = (float*)(ws + o); o += (size_t)NN * 256 * 4;      // out2 aliases
  float* ealpha = (float*)(ws + o); o += (size_t)(EE + NN) * 4 * 4; // layer2 aliases
  float* asrc   = (float*)(ws + o); o += (size_t)NN * 4 * 4;
  float* adst   = (float*)(ws + o); o += (size_t)NN * 4 * 4;
  unsigned* emax= (unsigned*)(ws + o); o += (size_t)NN * 4 * 4;
  float* denom  = (float*)(ws + o); o += (size_t)NN * 4 * 4;
  float* sums   = (float*)(ws + o); o += (size_t)NGRAPHS * HIDC * 4;
  float* cnts   = (float*)(ws + o); o += (size_t)NGRAPHS * 4;
  (void)ws_size; (void)in_sizes; (void)n_in; (void)out_size;

  const int TE1 = (EE + NN) * HEADS1;     // per-(edge,head) work, layer 1
  const int TE2 = (EE + NN);              // layer 2 (1 head)
  auto blocks = [](size_t n, int t) { return (int)((n + t - 1) / t); };

  // ================= layer 1 =================
  // GEMM1: h1[N,256] = x[N,1024] @ W1[1024,256]   (M=50000=3125*16 exact)
  wmma_gemm_f32_kernel<256, IN_DIM><<<dim3(NN / 16, 2), 128, 0, stream>>>(
      x, W1, h1);

  attn_node_kernel<<<blocks((size_t)NN * HEADS1, 256), 256, 0, stream>>>(
      h1, att_src1, att_dst1, asrc, adst, NN, HEADS1);

  zero_kernel<<<blocks((size_t)NN * HEADS1, 256), 256, 0, stream>>>((float*)emax, (size_t)NN * HEADS1);
  zero_kernel<<<blocks((size_t)NN * HEADS1, 256), 256, 0, stream>>>(denom, (size_t)NN * HEADS1);
  zero_kernel<<<blocks((size_t)NN * 256, 256), 256, 0, stream>>>(acc1, (size_t)NN * 256);

  edge_max_kernel<<<blocks(TE1, 256), 256, 0, stream>>>(asrc, adst, ei, emax, EE, NN, HEADS1);
  edge_exp_kernel<<<blocks(TE1, 256), 256, 0, stream>>>(asrc, adst, ei, emax, ealpha, denom, EE, NN, HEADS1);
  scatter_kernel<<<EE + NN, HEADS1 * HIDC, 0, stream>>>(h1, ealpha, denom, ei, acc1, EE, NN, HEADS1);
  bias_relu_kernel<<<blocks((size_t)NN * 256, 256), 256, 0, stream>>>(acc1, b1, 256, (size_t)NN * 256);

  // ================= layer 2 =================
  // GEMM2: h2[N,64] = acc1[N,256] @ W2[256,64]; h2 reuses h1 buffer
  float* h2 = h1;
  wmma_gemm_f32_kernel<64, 256><<<dim3(NN / 16, 1), 128, 0, stream>>>(
      acc1, W2, h2);

  attn_node_kernel<<<blocks(NN, 256), 256, 0, stream>>>(
      h2, att_src2, att_dst2, asrc, adst, NN, 1);

  float* acc2 = acc1;   // x2 fully consumed by GEMM2; reuse for accumulation
  zero_kernel<<<blocks(NN, 256), 256, 0, stream>>>((float*)emax, NN);
  zero_kernel<<<blocks(NN, 256), 256, 0, stream>>>(denom, NN);
  zero_kernel<<<blocks((size_t)NN * HIDC, 256), 256, 0, stream>>>(acc2, (size_t)NN * HIDC);

  edge_max_kernel<<<blocks(TE2, 256), 256, 0, stream>>>(asrc, adst, ei, emax, EE, NN, 1);
  edge_exp_kernel<<<blocks(TE2, 256), 256, 0, stream>>>(asrc, adst, ei, emax, ealpha, denom, EE, NN, 1);
  scatter_kernel<<<EE + NN, HIDC, 0, stream>>>(h2, ealpha, denom, ei, acc2, EE, NN, 1);
  bias_relu_kernel<<<blocks((size_t)NN * HIDC, 256), 256, 0, stream>>>(acc2, b2, HIDC, (size_t)NN * HIDC);

  // ================= mean pool + classifier =================
  zero_kernel<<<1, 256, 0, stream>>>(sums, NGRAPHS * HIDC + NGRAPHS); // sums+cnts contiguous
  pool_accum_kernel<<<blocks((size_t)NN * HIDC, 256), 256, 0, stream>>>(acc2, batch, sums, NN);
  pool_count_kernel<<<blocks(NN, 256), 256, 0, stream>>>(batch, cnts, NN);
  final_lin_kernel<<<1, 32, 0, stream>>>(sums, cnts, lin_w, lin_b, out);
}